// CDALayer_9929964388630
// MI455X (gfx1250) — compile-verified
//
#include <hip/hip_runtime.h>
#include <math.h>

typedef __attribute__((ext_vector_type(2))) float v2f;
typedef __attribute__((ext_vector_type(8))) float v8f;

#define BATCH 32
#define CH    256
#define CRR   16
#define HW    4096   // 64*64

// ---------------------------------------------------------------------------
// Kernel 1: global average pool  [32,256,64,64] -> y0 [32,256]
// One 256-thread block per (b,c); 4 x float4 per thread; tree-reduce in LDS.
// ---------------------------------------------------------------------------
__global__ void pool_kernel(const float* __restrict__ x, float* __restrict__ y0) {
  __shared__ float red[256];
  const int bc  = blockIdx.x;          // b*256 + c
  const int tid = threadIdx.x;
  const float4* xv = (const float4*)(x + (size_t)bc * HW);
  float s = 0.f;
#pragma unroll
  for (int i = 0; i < 4; ++i) {
    float4 v = xv[tid + i * 256];
    s += v.x + v.y + v.z + v.w;
  }
  red[tid] = s;
  __syncthreads();
  for (int off = 128; off > 0; off >>= 1) {
    if (tid < off) red[tid] += red[tid + off];
    __syncthreads();
  }
  if (tid == 0) y0[bc] = red[0] * (1.0f / (float)HW);
}

// ---------------------------------------------------------------------------
// f32 WMMA helper: D = A(16x4) * B(4x16) + C   (full fp32 precision path)
// ---------------------------------------------------------------------------
__device__ __forceinline__ v8f wmma4(v2f a, v2f b, v8f c) {
  return __builtin_amdgcn_wmma_f32_16x16x4_f32(
      /*neg_a=*/false, a, /*neg_b=*/false, b,
      /*c_mod=*/(short)0, c, /*reuse_a=*/false, /*reuse_b=*/false);
}

// ---------------------------------------------------------------------------
// Kernel 2: cross-linked MLP, single workgroup (8 wave32), all GEMMs via WMMA.
//   y1   = relu(y0 @ (w0_1+w01)^T)                       [32,16]
//   y2   = relu(y0 @ w02^T + y1 @ (w0_2+w12)^T)          [32,16]
//   gate = sigmoid(y0 @ w03^T + y1 @ w13^T + y2 @ (w0_3+w23)^T)  [32,256]
// Fragment layouts per CDNA5 ISA 7.12.2 (f32 16x16x4).
// ---------------------------------------------------------------------------
__global__ void __launch_bounds__(256, 1)
mlp_kernel(const float* __restrict__ y0,
           const float* __restrict__ w0_1, const float* __restrict__ w0_2,
           const float* __restrict__ w0_3, const float* __restrict__ w01,
           const float* __restrict__ w02,  const float* __restrict__ w03,
           const float* __restrict__ w12,  const float* __restrict__ w13,
           const float* __restrict__ w23,  float* __restrict__ gate) {
  __shared__ float ly1[BATCH * CRR];
  __shared__ float ly2[BATCH * CRR];
  const int tid  = threadIdx.x;
  const int wid  = tid >> 5;
  const int lane = tid & 31;
  const int lrow = lane & 15;   // M (for A) or N (for B/D) within fragment
  const int lhi  = lane >> 4;   // upper/lower half of wave

  // ---- Step 1: y1 = relu(y0 @ (w0_1+w01)^T) ----
  if (wid < 2) {
    const int m0 = wid * 16;
    v8f acc = {};
    for (int k0 = 0; k0 < CH; k0 += 4) {
      const int kk = k0 + 2 * lhi;
      v2f a, b;
      a.x = y0[(m0 + lrow) * CH + kk];
      a.y = y0[(m0 + lrow) * CH + kk + 1];
      b.x = w0_1[lrow * CH + kk]     + w01[lrow * CH + kk];
      b.y = w0_1[lrow * CH + kk + 1] + w01[lrow * CH + kk + 1];
      acc = wmma4(a, b, acc);
    }
#pragma unroll
    for (int j = 0; j < 8; ++j) {
      const int m = m0 + j + 8 * lhi;
      const float v = acc[j];
      ly1[m * CRR + lrow] = v > 0.f ? v : 0.f;
    }
  }
  __syncthreads();

  // ---- Step 2: y2 = relu(y0 @ w02^T + y1 @ (w0_2+w12)^T) ----
  if (wid < 2) {
    const int m0 = wid * 16;
    v8f acc = {};
    for (int k0 = 0; k0 < CH; k0 += 4) {
      const int kk = k0 + 2 * lhi;
      v2f a, b;
      a.x = y0[(m0 + lrow) * CH + kk];
      a.y = y0[(m0 + lrow) * CH + kk + 1];
      b.x = w02[lrow * CH + kk];
      b.y = w02[lrow * CH + kk + 1];
      acc = wmma4(a, b, acc);
    }
    for (int k0 = 0; k0 < CRR; k0 += 4) {
      const int kk = k0 + 2 * lhi;
      v2f a, b;
      a.x = ly1[(m0 + lrow) * CRR + kk];
      a.y = ly1[(m0 + lrow) * CRR + kk + 1];
      b.x = w0_2[lrow * CRR + kk]     + w12[lrow * CRR + kk];
      b.y = w0_2[lrow * CRR + kk + 1] + w12[lrow * CRR + kk + 1];
      acc = wmma4(a, b, acc);
    }
#pragma unroll
    for (int j = 0; j < 8; ++j) {
      const int m = m0 + j + 8 * lhi;
      const float v = acc[j];
      ly2[m * CRR + lrow] = v > 0.f ? v : 0.f;
    }
  }
  __syncthreads();

  // ---- Step 3: gate = sigmoid(y0@w03^T + y1@w13^T + y2@(w0_3+w23)^T) ----
  // 32 output tiles (2 M-tiles x 16 N-tiles) over 8 waves, 4 tiles each.
#pragma unroll
  for (int t = 0; t < 4; ++t) {
    const int tile = wid * 4 + t;
    const int m0 = (tile & 1) * 16;
    const int n0 = (tile >> 1) * 16;
    const int n  = n0 + lrow;
    v8f acc = {};
    for (int k0 = 0; k0 < CH; k0 += 4) {
      const int kk = k0 + 2 * lhi;
      v2f a, b;
      a.x = y0[(m0 + lrow) * CH + kk];
      a.y = y0[(m0 + lrow) * CH + kk + 1];
      b.x = w03[n * CH + kk];
      b.y = w03[n * CH + kk + 1];
      acc = wmma4(a, b, acc);
    }
    for (int k0 = 0; k0 < CRR; k0 += 4) {
      const int kk = k0 + 2 * lhi;
      v2f a, b;
      a.x = ly1[(m0 + lrow) * CRR + kk];
      a.y = ly1[(m0 + lrow) * CRR + kk + 1];
      b.x = w13[n * CRR + kk];
      b.y = w13[n * CRR + kk + 1];
      acc = wmma4(a, b, acc);
    }
    for (int k0 = 0; k0 < CRR; k0 += 4) {
      const int kk = k0 + 2 * lhi;
      v2f a, b;
      a.x = ly2[(m0 + lrow) * CRR + kk];
      a.y = ly2[(m0 + lrow) * CRR + kk + 1];
      b.x = w0_3[n * CRR + kk]     + w23[n * CRR + kk];
      b.y = w0_3[n * CRR + kk + 1] + w23[n * CRR + kk + 1];
      acc = wmma4(a, b, acc);
    }
#pragma unroll
    for (int j = 0; j < 8; ++j) {
      const int m = m0 + j + 8 * lhi;
      gate[m * CH + n] = 1.0f / (1.0f + __expf(-acc[j]));
    }
  }
}

// ---------------------------------------------------------------------------
// Kernel 3: out = x * gate[b,c]  (second x read should hit the 192MB L2)
// ---------------------------------------------------------------------------
__global__ void scale_kernel(const float* __restrict__ x,
                             const float* __restrict__ gate,
                             float* __restrict__ out) {
  const int bc  = blockIdx.x;
  const int tid = threadIdx.x;
  const float g = gate[bc];
  const float4* xv = (const float4*)(x + (size_t)bc * HW);
  float4* ov = (float4*)(out + (size_t)bc * HW);
#pragma unroll
  for (int i = 0; i < 4; ++i) {
    float4 v = xv[tid + i * 256];
    v.x *= g; v.y *= g; v.z *= g; v.w *= g;
    ov[tid + i * 256] = v;
  }
}

extern "C" void kernel_launch(void* const* d_in, const int* in_sizes, int n_in,
                              void* d_out, int out_size, void* d_ws, size_t ws_size,
                              hipStream_t stream) {
  const float* x    = (const float*)d_in[0];
  const float* w0_1 = (const float*)d_in[1];
  const float* w0_2 = (const float*)d_in[2];
  const float* w0_3 = (const float*)d_in[3];
  const float* w01  = (const float*)d_in[4];
  const float* w02  = (const float*)d_in[5];
  const float* w03  = (const float*)d_in[6];
  const float* w12  = (const float*)d_in[7];
  const float* w13  = (const float*)d_in[8];
  const float* w23  = (const float*)d_in[9];
  float* out  = (float*)d_out;
  float* y0   = (float*)d_ws;                // 32*256 f32
  float* gate = y0 + BATCH * CH;             // 32*256 f32

  pool_kernel<<<BATCH * CH, 256, 0, stream>>>(x, y0);
  mlp_kernel<<<1, 256, 0, stream>>>(y0, w0_1, w0_2, w0_3, w01, w02, w03,
                                    w12, w13, w23, gate);
  scale_kernel<<<BATCH * CH, 256, 0, stream>>>(x, gate, out);
}